// SVD_35966056137345
// MI455X (gfx1250) — compile-verified
//
#include <hip/hip_runtime.h>
#include <math.h>

typedef __attribute__((ext_vector_type(2))) float v2f;
typedef __attribute__((ext_vector_type(8))) float v8f;

constexpr int WAVES_PER_BLOCK = 2;                 // 64-thread blocks -> fine-grained WGP packing
constexpr int BLOCK_THREADS   = WAVES_PER_BLOCK * 32;
constexpr int ROWS_PER_BLOCK  = WAVES_PER_BLOCK * 16;

// ---------------------------------------------------------------------------
// Kernel 1: fused NN-search (WMMA f32 16x16x4, software-pipelined) +
//           per-point H/y/w + A,b reduction
// ---------------------------------------------------------------------------
__global__ void __launch_bounds__(BLOCK_THREADS)
icp_nn_accum_kernel(const float* __restrict__ src,
                    const float* __restrict__ oldtgt,   // (N,6): pts|nor
                    const float* __restrict__ flow,
                    const float* __restrict__ weight,
                    float* __restrict__ ws, int N)
{
    const int lane  = threadIdx.x & 31;
    const int wave  = threadIdx.x >> 5;
    const int half  = lane >> 4;          // 0: K=0,1   1: K=2,3 (pad)
    const int l     = lane & 15;
    const int batch = blockIdx.y;
    const int rowBase = blockIdx.x * ROWS_PER_BLOCK + wave * 16;

    const float* srcb  = src    + (size_t)batch * N * 3;
    const float* flowb = flow   + (size_t)batch * N * 3;
    const float* oldb  = oldtgt + (size_t)batch * N * 6;

    // ---- A fragment: 16x4 f32 (M = l, K split across halves, K=3 zero-pad)
    const int an = rowBase + l;
    const float tx = srcb[an * 3 + 0] + flowb[an * 3 + 0];
    const float ty = srcb[an * 3 + 1] + flowb[an * 3 + 1];
    const float tz = srcb[an * 3 + 2] + flowb[an * 3 + 2];
    v2f afrag;
    afrag.x = half ? tz   : tx;    // VGPR0: K=0 (lanes 0-15) / K=2 (lanes 16-31)
    afrag.y = half ? 0.0f : ty;    // VGPR1: K=1 / K=3 (pad)

    float bestScore[8];
    int   bestIdx[8];
#pragma unroll
    for (int v = 0; v < 8; ++v) { bestScore[v] = 3.402823466e+38f; bestIdx[v] = 0; }

    // ---- software-pipelined sweep, 2 candidate tiles (32 points) per step.
    //      score = |p|^2 - 2*(t.p)   (|t|^2 is constant per row -> dropped)
    float c0x = oldb[(size_t)(l)      * 6 + 0];
    float c0y = oldb[(size_t)(l)      * 6 + 1];
    float c0z = oldb[(size_t)(l)      * 6 + 2];
    float c1x = oldb[(size_t)(16 + l) * 6 + 0];
    float c1y = oldb[(size_t)(16 + l) * 6 + 1];
    float c1z = oldb[(size_t)(16 + l) * 6 + 2];

    for (int m0 = 0; m0 < N; m0 += 32) {
        // prefetch the next pair of tiles (wraps to 0 on last iter; discarded)
        int pb = m0 + 32;
        pb = (pb < N) ? pb : 0;
        const float n0x = oldb[(size_t)(pb + l)      * 6 + 0];
        const float n0y = oldb[(size_t)(pb + l)      * 6 + 1];
        const float n0z = oldb[(size_t)(pb + l)      * 6 + 2];
        const float n1x = oldb[(size_t)(pb + 16 + l) * 6 + 0];
        const float n1y = oldb[(size_t)(pb + 16 + l) * 6 + 1];
        const float n1z = oldb[(size_t)(pb + 16 + l) * 6 + 2];

        // ---- tile 0 (columns m0 .. m0+15)
        {
            const float pn2 = c0x * c0x + c0y * c0y + c0z * c0z;
            v2f bfrag;
            bfrag.x = half ? c0z  : c0x;
            bfrag.y = half ? 0.0f : c0y;
            v8f acc = {0.f, 0.f, 0.f, 0.f, 0.f, 0.f, 0.f, 0.f};
            acc = __builtin_amdgcn_wmma_f32_16x16x4_f32(false, afrag, false, bfrag,
                                                        (short)0, acc, false, false);
            const int m = m0 + l;  // lane holds column N=l for rows M = v + 8*half
#pragma unroll
            for (int v = 0; v < 8; ++v) {
                const float score  = __builtin_fmaf(-2.0f, acc[v], pn2);
                const bool  better = score < bestScore[v];
                bestIdx[v]   = better ? m     : bestIdx[v];
                bestScore[v] = better ? score : bestScore[v];
            }
        }
        // ---- tile 1 (columns m0+16 .. m0+31)
        {
            const float pn2 = c1x * c1x + c1y * c1y + c1z * c1z;
            v2f bfrag;
            bfrag.x = half ? c1z  : c1x;
            bfrag.y = half ? 0.0f : c1y;
            v8f acc = {0.f, 0.f, 0.f, 0.f, 0.f, 0.f, 0.f, 0.f};
            acc = __builtin_amdgcn_wmma_f32_16x16x4_f32(false, afrag, false, bfrag,
                                                        (short)0, acc, false, false);
            const int m = m0 + 16 + l;
#pragma unroll
            for (int v = 0; v < 8; ++v) {
                const float score  = __builtin_fmaf(-2.0f, acc[v], pn2);
                const bool  better = score < bestScore[v];
                bestIdx[v]   = better ? m     : bestIdx[v];
                bestScore[v] = better ? score : bestScore[v];
            }
        }
        // rotate the pipeline
        c0x = n0x; c0y = n0y; c0z = n0z;
        c1x = n1x; c1y = n1y; c1z = n1z;
    }

    // ---- argmin across the 16 column-lanes of each half (xor masks stay in-half)
#pragma unroll
    for (int v = 0; v < 8; ++v) {
#pragma unroll
        for (int mask = 8; mask >= 1; mask >>= 1) {
            const float os = __shfl_xor(bestScore[v], mask, 32);
            const int   oi = __shfl_xor(bestIdx[v],   mask, 32);
            const bool take = (os < bestScore[v]) ||
                              (os == bestScore[v] && oi < bestIdx[v]);
            bestScore[v] = take ? os : bestScore[v];
            bestIdx[v]   = take ? oi : bestIdx[v];
        }
    }

    // ---- distribute winning indices: rows 0-7 live in half0, rows 8-15 in half1
    __shared__ int s_idx[WAVES_PER_BLOCK][16];
    if (lane == 0) {
#pragma unroll
        for (int v = 0; v < 8; ++v) s_idx[wave][v] = bestIdx[v];
    } else if (lane == 16) {
#pragma unroll
        for (int v = 0; v < 8; ++v) s_idx[wave][8 + v] = bestIdx[v];
    }
    __syncthreads();

    // ---- per-row H, y, w and 6x6 normal-equation contributions
    float c27[27];
#pragma unroll
    for (int k = 0; k < 27; ++k) c27[k] = 0.0f;

    if (lane < 16) {
        const int n   = rowBase + lane;
        const int idx = s_idx[wave][lane];
        const float nx = oldb[(size_t)idx * 6 + 3];
        const float ny = oldb[(size_t)idx * 6 + 4];
        const float nz = oldb[(size_t)idx * 6 + 5];
        const float sx = srcb[n * 3 + 0];
        const float sy = srcb[n * 3 + 1];
        const float sz = srcb[n * 3 + 2];
        const float txx = sx + flowb[n * 3 + 0];
        const float tyy = sy + flowb[n * 3 + 1];
        const float tzz = sz + flowb[n * 3 + 2];
        const float y = (nx * txx + ny * tyy + nz * tzz)
                      - (nx * sx  + ny * sy  + nz * sz);
        const float H[6] = { nz * sy - ny * sz,
                             nx * sz - nz * sx,
                             ny * sx - nx * sy,
                             nx, ny, nz };
        const float w = weight[(size_t)batch * N + n];
        int k = 0;
#pragma unroll
        for (int i = 0; i < 6; ++i)
#pragma unroll
            for (int j = i; j < 6; ++j)
                c27[k++] = w * H[i] * H[j];
#pragma unroll
        for (int i = 0; i < 6; ++i) c27[21 + i] = w * H[i] * y;
    }

    // ---- wave reduction then one lane's 27 atomic adds per wave
#pragma unroll
    for (int k = 0; k < 27; ++k) {
#pragma unroll
        for (int mask = 16; mask >= 1; mask >>= 1)
            c27[k] += __shfl_xor(c27[k], mask, 32);
    }
    if (lane == 0) {
        float* accp = ws + batch * 32;
#pragma unroll
        for (int k = 0; k < 27; ++k) atomicAdd(&accp[k], c27[k]);
    }
}

// ---------------------------------------------------------------------------
// Kernel 0: zero the A/b accumulators (harness poisons ws; must re-zero)
// ---------------------------------------------------------------------------
__global__ void zero_ws_kernel(float* __restrict__ ws, int n)
{
    const int i = blockIdx.x * blockDim.x + threadIdx.x;
    if (i < n) ws[i] = 0.0f;
}

// ---------------------------------------------------------------------------
// Kernel 2: per-batch 6x6 solve (Gauss-Jordan, partial pivot) + Euler -> R, t
// ---------------------------------------------------------------------------
__global__ void solve_kernel(const float* __restrict__ ws,
                             float* __restrict__ out, int B)
{
    const int b = blockIdx.x * blockDim.x + threadIdx.x;
    if (b >= B) return;
    const float* p = ws + b * 32;

    float M[6][7];
    {
        float S[6][6];
        int k = 0;
        for (int i = 0; i < 6; ++i)
            for (int j = i; j < 6; ++j) { S[i][j] = p[k]; S[j][i] = p[k]; ++k; }
        for (int i = 0; i < 6; ++i) {
            for (int j = 0; j < 6; ++j) M[i][j] = S[i][j];
            M[i][6] = p[21 + i];
        }
    }
    for (int c = 0; c < 6; ++c) {
        int piv = c; float mx = fabsf(M[c][c]);
        for (int r = c + 1; r < 6; ++r) {
            const float v = fabsf(M[r][c]);
            if (v > mx) { mx = v; piv = r; }
        }
        if (piv != c)
            for (int j = 0; j < 7; ++j) { const float t = M[c][j]; M[c][j] = M[piv][j]; M[piv][j] = t; }
        const float d   = M[c][c];
        const float inv = (d != 0.0f) ? 1.0f / d : 0.0f;
        for (int r = 0; r < 6; ++r) {
            if (r == c) continue;
            const float f = M[r][c] * inv;
            for (int j = c; j < 7; ++j) M[r][j] -= f * M[c][j];
        }
    }
    float X[6];
    for (int c = 0; c < 6; ++c) {
        const float d = M[c][c];
        X[c] = (d != 0.0f) ? M[c][6] / d : 0.0f;
    }

    const float ca = cosf(X[0]), sa = sinf(X[0]);
    const float cb = cosf(X[1]), sb = sinf(X[1]);
    const float cc = cosf(X[2]), sc = sinf(X[2]);
    float R[9];
    R[0] = cc * cb;  R[1] = cc * sb * sa - sc * ca;  R[2] = cc * sb * ca + sc * sa;
    R[3] = sc * cb;  R[4] = sc * sb * sa + cc * ca;  R[5] = sc * sb * ca - cc * sa;
    R[6] = -sb;      R[7] = cb * sa;                 R[8] = cb * ca;

    for (int i = 0; i < 9; ++i) out[b * 9 + i] = R[i];             // R: (B,3,3) flat
    for (int i = 0; i < 3; ++i) out[B * 9 + b * 3 + i] = X[3 + i]; // t: (B,3) flat
}

// ---------------------------------------------------------------------------
extern "C" void kernel_launch(void* const* d_in, const int* in_sizes, int n_in,
                              void* d_out, int out_size, void* d_ws, size_t ws_size,
                              hipStream_t stream)
{
    (void)n_in; (void)out_size; (void)ws_size;
    const float* src    = (const float*)d_in[0];
    const float* oldtgt = (const float*)d_in[1];
    // d_in[2] (feat2_l0_1) is unused by the reference
    const float* flow   = (const float*)d_in[3];
    const float* weight = (const float*)d_in[4];
    float* out = (float*)d_out;
    float* ws  = (float*)d_ws;

    const int N = 8192;
    const int B = in_sizes[0] / (N * 3);

    zero_ws_kernel<<<1, 64, 0, stream>>>(ws, B * 32);
    dim3 grid(N / ROWS_PER_BLOCK, B);
    icp_nn_accum_kernel<<<grid, BLOCK_THREADS, 0, stream>>>(src, oldtgt, flow, weight, ws, N);
    solve_kernel<<<1, 32, 0, stream>>>(ws, out, B);
}